// DomainInvariantFeaturesLearningNetwork_26749056319849
// MI455X (gfx1250) — compile-verified
//
#include <hip/hip_runtime.h>
#include <hip/hip_bf16.h>
#include <math.h>

// ---------------- problem constants ----------------
#define FEAT_DIM 2048
#define HID      256
#define NPTS     1024
#define BN_EPS   1e-5f

typedef __attribute__((ext_vector_type(2))) float v2f;
typedef __attribute__((ext_vector_type(8))) float v8f;

// =====================================================================
// WMMA f32 GEMM:  C[M][Nc] = A[M][K] @ B[K][Nc] (+ bias per column)
// Row-major, lda=K, ldb=Nc, ldc=Nc.
// Each wave32 computes a 16x64 output tile: one A fragment is reused by
// 4 B fragments -> 4x V_WMMA_F32_16X16X4_F32 per k-step. The inner loop
// uses pure pointer increments (no per-iteration index multiplies).
// Block = 4 waves covering 4 adjacent M tiles (B fragments shared via WGP$).
// Requires: M % 64 == 0, Nc % 64 == 0, K % 4 == 0 (all hold here).
// =====================================================================
__global__ __launch_bounds__(128)
void gemm_f32_wmma(const float* __restrict__ A, const float* __restrict__ B,
                   const float* __restrict__ bias, float* __restrict__ C,
                   int M, int Nc, int K) {
  const int lane = threadIdx.x;                    // 0..31, all active
  const int half = lane >> 4;                      // selects K{0,1}/{2,3}, M row half
  const int l    = lane & 15;
  const int m0   = (blockIdx.x * 4 + threadIdx.y) * 16;
  const int n0   = blockIdx.y * 64;
  const int koff = 2 * half;

  const float* __restrict__ Ap = A + (size_t)(m0 + l) * K + koff;
  const float* __restrict__ Bp = B + (size_t)koff * Nc + n0 + l;
  const size_t bstep = (size_t)4 * Nc;

  v8f acc0 = {}, acc1 = {}, acc2 = {}, acc3 = {};

  #pragma unroll 4
  for (int k = 0; k < K; k += 4) {
    if (((k & 63) == 0) && (k + 256) < K) {
      __builtin_prefetch(Ap + 256, 0, 3);          // near-scope global_prefetch_b8
    }
    v2f a;
    a.x = Ap[0];                                   // contiguous pair -> b64 load
    a.y = Ap[1];
    v2f b0, b1, b2, b3;
    b0.x = Bp[0];        b0.y = Bp[Nc + 0];
    b1.x = Bp[16];       b1.y = Bp[Nc + 16];
    b2.x = Bp[32];       b2.y = Bp[Nc + 32];
    b3.x = Bp[48];       b3.y = Bp[Nc + 48];
    // 8 args: (neg_a, A, neg_b, B, c_mod, C, reuse_a, reuse_b)
    acc0 = __builtin_amdgcn_wmma_f32_16x16x4_f32(false, a, false, b0, (short)0, acc0, false, false);
    acc1 = __builtin_amdgcn_wmma_f32_16x16x4_f32(false, a, false, b1, (short)0, acc1, false, false);
    acc2 = __builtin_amdgcn_wmma_f32_16x16x4_f32(false, a, false, b2, (short)0, acc2, false, false);
    acc3 = __builtin_amdgcn_wmma_f32_16x16x4_f32(false, a, false, b3, (short)0, acc3, false, false);
    Ap += 4;
    Bp += bstep;
  }

  // epilogue: C/D layout -> lane half selects M+0 / M+8; VGPR r = row offset
  float* __restrict__ Cout = C + (size_t)(m0 + half * 8) * Nc + n0 + l;
  const float bv0 = bias ? bias[n0 + l +  0] : 0.0f;
  const float bv1 = bias ? bias[n0 + l + 16] : 0.0f;
  const float bv2 = bias ? bias[n0 + l + 32] : 0.0f;
  const float bv3 = bias ? bias[n0 + l + 48] : 0.0f;
  #pragma unroll
  for (int r = 0; r < 8; ++r) {
    float* row = Cout + (size_t)r * Nc;
    row[ 0] = acc0[r] + bv0;
    row[16] = acc1[r] + bv1;
    row[32] = acc2[r] + bv2;
    row[48] = acc3[r] + bv3;
  }
}

// =====================================================================
// BatchNorm statistics: per-column sum and sum-of-squares over N rows.
// Block b handles an 8-row slab; thread t owns column t (coalesced).
// =====================================================================
__global__ __launch_bounds__(HID)
void col_stats_kernel(const float* __restrict__ Y,
                      float* __restrict__ colsum,
                      float* __restrict__ colsumsq) {
  const int col = threadIdx.x;
  const int r0  = blockIdx.x * 8;
  float s = 0.f, s2 = 0.f;
  #pragma unroll
  for (int r = 0; r < 8; ++r) {
    const float v = Y[(size_t)(r0 + r) * HID + col];
    s  += v;
    s2 += v * v;
  }
  atomicAdd(&colsum[col],   s);
  atomicAdd(&colsumsq[col], s2);
}

__global__ __launch_bounds__(256)
void zero_kernel(float* __restrict__ p, int n) {
  const int i = blockIdx.x * 256 + threadIdx.x;
  if (i < n) p[i] = 0.f;
}

// BN (training-mode, biased variance) + ReLU
__global__ __launch_bounds__(256)
void bn_relu_kernel(const float* __restrict__ Y,
                    const float* __restrict__ colsum,
                    const float* __restrict__ colsumsq,
                    const float* __restrict__ g,
                    const float* __restrict__ bt,
                    float* __restrict__ out) {
  const int idx = blockIdx.x * 256 + threadIdx.x;
  const int col = idx & (HID - 1);
  const float invN = 1.0f / (float)NPTS;
  const float mean = colsum[col] * invN;
  const float var  = colsumsq[col] * invN - mean * mean;
  const float inv  = rsqrtf(var + BN_EPS);
  const float v    = (Y[idx] - mean) * inv * g[col] + bt[col];
  out[idx] = fmaxf(v, 0.f);
}

// =====================================================================
// Transpose hj [NPTS][HID] -> hjT [HID][NPTS] via LDS tile (conflict-free)
// =====================================================================
__global__ __launch_bounds__(256)
void transpose_kernel(const float* __restrict__ src, float* __restrict__ dst) {
  __shared__ float tile[32][33];
  const int tx = threadIdx.x;   // 0..31
  const int ty = threadIdx.y;   // 0..7
  const int bj = blockIdx.x;    // row-block of src  (NPTS/32 = 32)
  const int bh = blockIdx.y;    // col-block of src  (HID/32  = 8)
  #pragma unroll
  for (int r = 0; r < 4; ++r) {
    tile[ty + r * 8][tx] = src[(size_t)(bj * 32 + ty + r * 8) * HID + bh * 32 + tx];
  }
  __syncthreads();
  #pragma unroll
  for (int r = 0; r < 4; ++r) {
    dst[(size_t)(bh * 32 + ty + r * 8) * NPTS + bj * 32 + tx] = tile[tx][ty + r * 8];
  }
}

// =====================================================================
// All-pairs edge weights. ReLU is inside the H-reduction (not a GEMM).
// E[i][j] = mask(i,j) * sigmoid( sum_h relu(hi[i,h]+hjT[h,j]) * we2[h] + bwe2 )
// hi already has bwe1 folded in (GEMM3 bias). hi/we2 reads are wave-uniform
// (scalarized); hjT reads are fully coalesced over j.
// =====================================================================
__global__ __launch_bounds__(256)
void pair_edge_kernel(const float* __restrict__ hi,
                      const float* __restrict__ hjT,
                      const float* __restrict__ we2,
                      const float* __restrict__ bwe2,
                      const int*   __restrict__ labels,
                      float* __restrict__ E) {
  const int i = blockIdx.x;
  const int j = blockIdx.y * 256 + threadIdx.x;
  const float* __restrict__ hrow = hi + (size_t)i * HID;

  float acc = 0.f;
  #pragma unroll 8
  for (int h = 0; h < HID; ++h) {
    const float s = hrow[h] + hjT[(size_t)h * NPTS + j];
    acc = fmaf(fmaxf(s, 0.f), we2[h], acc);
  }
  acc += bwe2[0];

  float w = 0.f;
  if ((labels[i] == labels[j]) && (i != j)) {
    w = 1.0f / (1.0f + __expf(-acc));
  }
  E[(size_t)i * NPTS + j] = w;
}

// Row sums of E -> wsum[i]
__global__ __launch_bounds__(256)
void row_sum_kernel(const float* __restrict__ E, float* __restrict__ wsum) {
  __shared__ float sm[256];
  const int i = blockIdx.x;
  float s = 0.f;
  for (int j = threadIdx.x; j < NPTS; j += 256) s += E[(size_t)i * NPTS + j];
  sm[threadIdx.x] = s;
  __syncthreads();
  for (int off = 128; off > 0; off >>= 1) {
    if (threadIdx.x < off) sm[threadIdx.x] += sm[threadIdx.x + off];
    __syncthreads();
  }
  if (threadIdx.x == 0) wsum[i] = sm[0];
}

// updated = di + (wsum>0 ? weighted_sum/wsum : 0)
__global__ __launch_bounds__(256)
void final_update_kernel(const float* __restrict__ di,
                         const float* __restrict__ wsmat,
                         const float* __restrict__ wsum,
                         float* __restrict__ out) {
  const int idx = blockIdx.x * 256 + threadIdx.x;
  const int i = idx >> 8;  // HID == 256
  const float w = wsum[i];
  const float add = (w > 0.f) ? (wsmat[idx] / w) : 0.f;
  out[idx] = di[idx] + add;
}

// =====================================================================
// Host-side orchestration
// =====================================================================
extern "C" void kernel_launch(void* const* d_in, const int* in_sizes, int n_in,
                              void* d_out, int out_size, void* d_ws, size_t ws_size,
                              hipStream_t stream) {
  (void)in_sizes; (void)n_in; (void)out_size; (void)ws_size;

  const float* features = (const float*)d_in[0];   // [N, F]
  const int*   labels   = (const int*)  d_in[1];   // [N]
  const float* W1  = (const float*)d_in[2];        // [F, H]
  const float* b1  = (const float*)d_in[3];
  const float* g1  = (const float*)d_in[4];
  const float* bt1 = (const float*)d_in[5];
  const float* W2  = (const float*)d_in[6];        // [H, H]
  const float* b2  = (const float*)d_in[7];
  const float* g2  = (const float*)d_in[8];
  const float* bt2 = (const float*)d_in[9];
  const float* We1 = (const float*)d_in[10];       // [2H, H]
  const float* bwe1= (const float*)d_in[11];       // [H]
  const float* We2 = (const float*)d_in[12];       // [H, 1] (contiguous vec)
  const float* bwe2= (const float*)d_in[13];       // [1]
  float* out = (float*)d_out;                      // [N, H]

  // workspace layout (floats)
  float* ws = (float*)d_ws;
  const size_t MH = (size_t)NPTS * HID;            // 262144
  float* Y     = ws;                // GEMM scratch (reused)
  float* h     = ws + 1 * MH;
  float* di    = ws + 2 * MH;
  float* hi    = ws + 3 * MH;      // bwe1 folded in
  float* hj    = ws + 4 * MH;
  float* hjT   = ws + 5 * MH;      // [H][N]
  float* wsmat = ws + 6 * MH;      // weighted_sum
  float* E     = ws + 7 * MH;      // [N][N] = 1M floats
  float* colsum   = ws + 7 * MH + (size_t)NPTS * NPTS;
  float* colsumsq = colsum + HID;
  float* wsum     = colsumsq + HID;                // [N]

  const dim3 gblock(32, 4, 1);                      // 4 waves / block
  const dim3 gtiles(NPTS / 64, HID / 64, 1);        // 16 x 4 blocks (16x64 tile/wave)

  // ---- stage 1: Linear(F->H) + BN + ReLU ----
  zero_kernel<<<dim3(2), dim3(256), 0, stream>>>(colsum, 2 * HID);
  gemm_f32_wmma<<<gtiles, gblock, 0, stream>>>(features, W1, b1, Y,
                                               NPTS, HID, FEAT_DIM);
  col_stats_kernel<<<dim3(NPTS / 8), dim3(HID), 0, stream>>>(Y, colsum, colsumsq);
  bn_relu_kernel<<<dim3(MH / 256), dim3(256), 0, stream>>>(Y, colsum, colsumsq,
                                                           g1, bt1, h);

  // ---- stage 2: Linear(H->H) + BN + ReLU ----
  zero_kernel<<<dim3(2), dim3(256), 0, stream>>>(colsum, 2 * HID);
  gemm_f32_wmma<<<gtiles, gblock, 0, stream>>>(h, W2, b2, Y,
                                               NPTS, HID, HID);
  col_stats_kernel<<<dim3(NPTS / 8), dim3(HID), 0, stream>>>(Y, colsum, colsumsq);
  bn_relu_kernel<<<dim3(MH / 256), dim3(256), 0, stream>>>(Y, colsum, colsumsq,
                                                           g2, bt2, di);

  // ---- edge network preamble: hi = di@We1[:H] + bwe1 ; hj = di@We1[H:] ----
  gemm_f32_wmma<<<gtiles, gblock, 0, stream>>>(di, We1, bwe1, hi,
                                               NPTS, HID, HID);
  gemm_f32_wmma<<<gtiles, gblock, 0, stream>>>(di, We1 + (size_t)HID * HID,
                                               nullptr, hj, NPTS, HID, HID);
  transpose_kernel<<<dim3(NPTS / 32, HID / 32), dim3(32, 8), 0, stream>>>(hj, hjT);

  // ---- all-pairs edge weights ----
  pair_edge_kernel<<<dim3(NPTS, NPTS / 256), dim3(256), 0, stream>>>(
      hi, hjT, We2, bwe2, labels, E);
  row_sum_kernel<<<dim3(NPTS), dim3(256), 0, stream>>>(E, wsum);

  // ---- aggregation: weighted_sum = E @ di  (WMMA GEMM, K = N) ----
  gemm_f32_wmma<<<gtiles, gblock, 0, stream>>>(E, di, nullptr, wsmat,
                                               NPTS, HID, NPTS);

  // ---- final node update ----
  final_update_kernel<<<dim3(MH / 256), dim3(256), 0, stream>>>(di, wsmat, wsum, out);
}